// CliffordISTA_33243046871242
// MI455X (gfx1250) — compile-verified
//
#include <hip/hip_runtime.h>
#include <math.h>

// ---------------------------------------------------------------------------
// Clifford ISTA on MI455X (gfx1250), fp32 WMMA (v_wmma_f32_16x16x4_f32).
//
// Shapes: B=64, M=256, N=512, nb=8  ->  K_fwd [4096,2048], K_bwd [2048,4096]
// Per iter: err = x @ K_fwd - y ; grad = err @ K_bwd ; x = ST(x - 0.01*grad)
//
// K operators are stored K-PACKED: element (krow, col) of the logical matrix
// lives at ((krow>>2)*ND + col)*4 + (krow&3).  A WMMA k-step (4 rows) for one
// column tile is then a single contiguous 256B span per wave, loaded as one
// b64 per lane (lane half h takes k = 2h, 2h+1).
// ---------------------------------------------------------------------------

typedef float v2f __attribute__((ext_vector_type(2)));
typedef float v8f __attribute__((ext_vector_type(8)));

#define NB        8
#define BATCH     64
#define KF_ROWS   4096              // N*nb
#define KF_COLS   2048              // M*nb
#define KF_ELEMS  (KF_ROWS * KF_COLS)   // 8388608 (same count for K_bwd)
#define WAVES     8
#define TPB       (WAVES * 32)
#define N_ITER    50
#define STEP      0.01f

// per-blade soft threshold: LAMBDAS[grade(blade)] with grades {0,1,1,2,1,2,2,3}
__constant__ float c_thr[8] = {0.0f, 0.001f, 0.001f, 0.001f,
                               0.001f, 0.001f, 0.001f, 0.002f};

// Cayley reorder sign for e_a * e_b in Euclidean Cl(3,0)
__device__ __forceinline__ float cayley_sign(int a, int b) {
    int cnt = 0;
    int aa = a >> 1;
    while (aa) { cnt += __popc(aa & b); aa >>= 1; }
    return (cnt & 1) ? -1.0f : 1.0f;
}

// K_fwd[(n*8+i),(m*8+k)] = sign(i, i^k) * A[m,n,i^k]   (written k-packed)
__global__ void cista_build_kfwd(const float* __restrict__ A, float* __restrict__ Kf) {
    int idx = blockIdx.x * blockDim.x + threadIdx.x;   // packed index
    int blk  = idx >> 13;            // / (KF_COLS*4)
    int rem  = idx & 8191;
    int col  = rem >> 2;
    int krow = (blk << 2) | (rem & 3);
    int n = krow >> 3, i = krow & 7;
    int m = col  >> 3, k = col  & 7;
    int j = i ^ k;
    Kf[idx] = cayley_sign(i, j) * A[((m << 9) + n) * NB + j];
}

// K_bwd[(m*8+i),(n*8+k)] = rev(i^k) * sign(i, i^k) * A[m,n,i^k]  (k-packed)
__global__ void cista_build_kbwd(const float* __restrict__ A, float* __restrict__ Kb) {
    int idx = blockIdx.x * blockDim.x + threadIdx.x;
    int blk  = idx >> 14;            // / (KF_ROWS*4)
    int rem  = idx & 16383;
    int col  = rem >> 2;
    int krow = (blk << 2) | (rem & 3);
    int m = krow >> 3, i = krow & 7;
    int n = col  >> 3, k = col  & 7;
    int j = i ^ k;
    int g = __popc(j);                         // grade; rev = (-1)^(g(g-1)/2)
    float rev = ((g >> 1) & 1) ? -1.0f : 1.0f; // g in {0..3}: + + - -
    Kb[idx] = rev * cayley_sign(i, j) * A[((m << 9) + n) * NB + j];
}

__global__ void cista_zero_x(float* __restrict__ x) {
    x[blockIdx.x * blockDim.x + threadIdx.x] = 0.0f;
}

// ---------------------------------------------------------------------------
// 64 x 16 output tile per block via v_wmma_f32_16x16x4_f32.
// 8 waves split the K dimension (scalar, wave-uniform loop bounds); partial
// tiles reduced through LDS.
//
// Fragment layouts (ISA 7.12.2):
//   A (16x4 f32):  lane L holds M = L%16, VGPR v holds K = v + 2*(L/16)
//   B (4x16 f32):  lane L holds N = L%16, VGPR v holds K = v + 2*(L/16)
//   C/D (16x16):   lane L holds N = L%16, VGPR v holds M = v + 8*(L/16)
// ---------------------------------------------------------------------------
template <int KD, int ND>
__device__ __forceinline__ void gemm_tile_wmma(const float* __restrict__ X,
                                               const float* __restrict__ Kpk,
                                               int col0, float* smem) {
    const int tid  = threadIdx.x;
    const int lane = tid & 31;
    const int half = lane >> 4;
    const int lc   = lane & 15;
    // wave index is uniform across the wave -> force scalar loop bounds
    const int wave = __builtin_amdgcn_readfirstlane(tid >> 5);
    constexpr int KCHUNK = KD / WAVES;
    const int k0 = wave * KCHUNK;

    v8f acc0 = {0.f, 0.f, 0.f, 0.f, 0.f, 0.f, 0.f, 0.f};
    v8f acc1 = acc0, acc2 = acc0, acc3 = acc0;

    const float* xb = X + (size_t)lc * KD + 2 * half;          // row = mt*16+lc
    const float* kb = Kpk + ((size_t)(col0 + lc) << 2) + 2 * half;

#pragma unroll 2
    for (int kk = k0; kk < k0 + KCHUNK; kk += 4) {
        // B fragment: one b64 per lane, wave covers contiguous 256B
        v2f bfrag = *(const v2f*)(kb + (size_t)kk * ND);       // (kk>>2)*(ND*4)
        v2f a0 = *(const v2f*)(xb + (size_t)0 * 16 * KD + kk);
        v2f a1 = *(const v2f*)(xb + (size_t)1 * 16 * KD + kk);
        v2f a2 = *(const v2f*)(xb + (size_t)2 * 16 * KD + kk);
        v2f a3 = *(const v2f*)(xb + (size_t)3 * 16 * KD + kk);
        acc0 = __builtin_amdgcn_wmma_f32_16x16x4_f32(false, a0, false, bfrag,
                                                     (short)0, acc0, false, false);
        acc1 = __builtin_amdgcn_wmma_f32_16x16x4_f32(false, a1, false, bfrag,
                                                     (short)0, acc1, false, false);
        acc2 = __builtin_amdgcn_wmma_f32_16x16x4_f32(false, a2, false, bfrag,
                                                     (short)0, acc2, false, false);
        acc3 = __builtin_amdgcn_wmma_f32_16x16x4_f32(false, a3, false, bfrag,
                                                     (short)0, acc3, false, false);
    }

    // scatter C/D fragments to LDS: smem[wave][row(0..63)][col(0..15)]
    float* sw = smem + wave * 1024;
#pragma unroll
    for (int v = 0; v < 8; ++v) {
        const int rb = v + 8 * half;
        sw[(rb +  0) * 16 + lc] = acc0[v];
        sw[(rb + 16) * 16 + lc] = acc1[v];
        sw[(rb + 32) * 16 + lc] = acc2[v];
        sw[(rb + 48) * 16 + lc] = acc3[v];
    }
    __syncthreads();
}

// err = x @ K_fwd - y        (x: [64,4096], K_fwd: [4096,2048], err/y: [64,2048])
__global__ void __launch_bounds__(TPB)
cista_fwd(const float* __restrict__ x, const float* __restrict__ Kf,
          const float* __restrict__ y, float* __restrict__ err) {
    __shared__ float smem[WAVES * 1024];
    const int col0 = blockIdx.x << 4;
    gemm_tile_wmma<KF_ROWS, KF_COLS>(x, Kf, col0, smem);

    for (int e = threadIdx.x; e < 1024; e += TPB) {
        float s = 0.f;
#pragma unroll
        for (int w = 0; w < WAVES; ++w) s += smem[w * 1024 + e];
        const int r = e >> 4, c = e & 15;
        const size_t gi = (size_t)r * KF_COLS + col0 + c;
        err[gi] = s - y[gi];
    }
}

// grad = err @ K_bwd ; x = softthresh(x - STEP*grad)   (x: [64,4096] in d_out)
__global__ void __launch_bounds__(TPB)
cista_bwd(const float* __restrict__ err, const float* __restrict__ Kb,
          float* __restrict__ x) {
    __shared__ float smem[WAVES * 1024];
    const int col0 = blockIdx.x << 4;
    gemm_tile_wmma<KF_COLS, KF_ROWS>(err, Kb, col0, smem);

    for (int e = threadIdx.x; e < 1024; e += TPB) {
        float g = 0.f;
#pragma unroll
        for (int w = 0; w < WAVES; ++w) g += smem[w * 1024 + e];
        const int r = e >> 4, c = e & 15;
        const int col = col0 + c;
        const size_t gi = (size_t)r * KF_ROWS + col;
        const float xv = x[gi] - STEP * g;
        const float mag = fabsf(xv) - c_thr[col & 7];
        x[gi] = (mag > 0.f) ? copysignf(mag, xv) : 0.f;
    }
}

extern "C" void kernel_launch(void* const* d_in, const int* in_sizes, int n_in,
                              void* d_out, int out_size, void* d_ws, size_t ws_size,
                              hipStream_t stream) {
    (void)in_sizes; (void)n_in; (void)out_size; (void)ws_size;

    const float* y = (const float*)d_in[0];   // [64,256,8]
    const float* A = (const float*)d_in[1];   // [256,512,8]
    float* x  = (float*)d_out;                // [64,512,8] -> x lives here
    float* Kf = (float*)d_ws;                 // 8388608 f32 (32 MB), k-packed
    float* Kb = Kf + KF_ELEMS;                // 8388608 f32 (32 MB), k-packed
    float* er = Kb + KF_ELEMS;                // 131072 f32

    // Build fused Clifford operators once (amortized over 50 iterations).
    cista_build_kfwd<<<KF_ELEMS / 256, 256, 0, stream>>>(A, Kf);
    cista_build_kbwd<<<KF_ELEMS / 256, 256, 0, stream>>>(A, Kb);
    cista_zero_x<<<(BATCH * KF_ROWS) / 256, 256, 0, stream>>>(x);

    for (int it = 0; it < N_ITER; ++it) {
        cista_fwd<<<KF_COLS / 16, TPB, 0, stream>>>(x, Kf, y, er);
        cista_bwd<<<KF_ROWS / 16, TPB, 0, stream>>>(er, Kb, x);
    }
}